// RADAMPooling_63986422776031
// MI455X (gfx1250) — compile-verified
//
#include <hip/hip_runtime.h>
#include <math.h>

#define B_ 32
#define C_ 2048
#define S_ 28
#define N_ (S_*S_)      /* 784 */
#define M_ 4
#define LAMBDA_ 1e-3f
#define EPS_ 1e-10f
#define LN1E4_ 9.210340371976184f   /* ln(10000) */
#define HALFPI_ 1.5707963267948966f

typedef __attribute__((ext_vector_type(2))) float v2f;
typedef __attribute__((ext_vector_type(8))) float v8f;

__device__ __forceinline__ float waveReduceSum(float v) {
  #pragma unroll
  for (int off = 16; off > 0; off >>= 1)
    v += __shfl_xor(v, off, 32);
  return v;
}

// ---------------------------------------------------------------------------
// Kernel 1: per-(b,c) L2 norm over the 784 spatial elements.
// One wave per channel map; contiguous 3136-byte reads per wave.
// ---------------------------------------------------------------------------
__global__ void __launch_bounds__(256) norm_kernel(const float* __restrict__ x,
                                                   float* __restrict__ invn) {
  int wave = (blockIdx.x * blockDim.x + threadIdx.x) >> 5;   // 0 .. B*C-1 (exact)
  int lane = threadIdx.x & 31;
  const float* p = x + (size_t)wave * N_;
  float s = 0.f;
  for (int i = lane; i < N_; i += 32) { float v = p[i]; s += v * v; }
  s = waveReduceSum(s);
  if (lane == 0) invn[wave] = 1.0f / fmaxf(sqrtf(s), EPS_);
}

// ---------------------------------------------------------------------------
// Kernel 2: hid[b,n,m] = sigmoid( sum_c (x[b,c,n]*invn[b,c] + pe[n,c]) * W[m,c] )
// Block = 256 threads = one batch, 256 consecutive n. W (32KB) and invn (8KB)
// staged in LDS; x loads coalesced across lanes (consecutive n). PE recomputed
// via one __sincosf per channel pair (uniform, no divergence).
// ---------------------------------------------------------------------------
__global__ void __launch_bounds__(256) hid_kernel(const float* __restrict__ x,
                                                  const float* __restrict__ W,
                                                  const float* __restrict__ invn,
                                                  float* __restrict__ hid) {
  __shared__ float Wl[C_ * M_];   // [c][m]
  __shared__ float nl[C_];
  const int b    = blockIdx.x;
  const int tile = blockIdx.y;
  const int tid  = threadIdx.x;

  for (int idx = tid; idx < C_ * M_; idx += 256) {
    int c = idx >> 2, m = idx & 3;
    Wl[idx] = W[m * C_ + c];      // W is (M, C, 1)
  }
  for (int c = tid; c < C_; c += 256) nl[c] = invn[b * C_ + c];
  __syncthreads();

  const int  n     = tile * 256 + tid;
  const bool valid = (n < N_);
  const int  nn    = valid ? n : 0;
  const float fn   = (float)nn;
  const float* xb  = x + (size_t)b * C_ * N_ + nn;
  const float kk   = LN1E4_ / (float)C_;

  float a0 = 0.f, a1 = 0.f, a2 = 0.f, a3 = 0.f;
  for (int c = 0; c < C_; c += 2) {
    float x0  = xb[(size_t)c * N_];
    float x1  = xb[(size_t)(c + 1) * N_];
    float div = __expf(-(float)c * kk);
    float sv, cv;
    __sincosf(fn * div, &sv, &cv);
    float e0 = x0 * nl[c]     + sv;
    float e1 = x1 * nl[c + 1] + cv;
    a0 += e0 * Wl[c * 4 + 0] + e1 * Wl[(c + 1) * 4 + 0];
    a1 += e0 * Wl[c * 4 + 1] + e1 * Wl[(c + 1) * 4 + 1];
    a2 += e0 * Wl[c * 4 + 2] + e1 * Wl[(c + 1) * 4 + 2];
    a3 += e0 * Wl[c * 4 + 3] + e1 * Wl[(c + 1) * 4 + 3];
  }
  if (valid) {
    float4 h;
    h.x = 1.f / (1.f + __expf(-a0));
    h.y = 1.f / (1.f + __expf(-a1));
    h.z = 1.f / (1.f + __expf(-a2));
    h.w = 1.f / (1.f + __expf(-a3));
    *(float4*)(hid + ((size_t)b * N_ + n) * M_) = h;
  }
}

// ---------------------------------------------------------------------------
// Kernel 3: hth[b,m] = sum_n hid[b,n,m]^2.  One wave per (b,m); tiny.
// ---------------------------------------------------------------------------
__global__ void __launch_bounds__(32) hth_kernel(const float* __restrict__ hid,
                                                 float* __restrict__ hth) {
  int bm = blockIdx.x;              // 0 .. B*M-1
  int b = bm >> 2, m = bm & 3;
  int lane = threadIdx.x;
  const float* hb = hid + (size_t)b * N_ * M_ + m;
  float s = 0.f;
  for (int nn = lane; nn < N_; nn += 32) { float h = hb[(size_t)nn * M_]; s += h * h; }
  s = waveReduceSum(s);
  if (lane == 0) hth[bm] = s;
}

// ---------------------------------------------------------------------------
// Kernel 4: per batch, htx(16pad x 16ctile) += hid^T(16x4) @ xp(4x16) via
// V_WMMA_F32_16X16X4_F32, K stepped over n in chunks of 4 (196 WMMAs/wave).
// A layout (16x4 f32): lanes 0-15 carry K=0,1 in v0,v1; lanes 16-31 carry K=2,3.
// B/C/D row-striped across lanes. PE uses cos(x)=sin(x+pi/2) so the per-lane
// even/odd channel choice is a constant phase add, NOT a divergent branch.
// Then beta = acc[m]/(hth+lambda), sum over m, nan_to_num, store pooled[b,c].
// ---------------------------------------------------------------------------
__global__ void __launch_bounds__(32) htx_wmma_kernel(const float* __restrict__ x,
                                                      const float* __restrict__ invn,
                                                      const float* __restrict__ hid,
                                                      const float* __restrict__ hth,
                                                      float* __restrict__ out) {
  const int ct   = blockIdx.x;          // channel tile, 0..127
  const int b    = blockIdx.y;          // batch
  const int lane = threadIdx.x;         // 0..31 (wave32)
  const int L    = lane & 15;
  const int kOff = (lane >> 4) << 1;    // 0 for lanes 0-15, 2 for lanes 16-31
  const int c    = ct * 16 + L;

  // Per-lane loop invariants for xp recomputation.
  const float invc  = invn[b * C_ + c];
  const float kk    = LN1E4_ / (float)C_;
  const float div   = __expf(-(float)(c & ~1) * kk);
  const float phase = ((c & 1) == 0) ? 0.f : HALFPI_;   // sin vs cos, branch-free
  const float* xrow = x + ((size_t)b * C_ + c) * N_;
  const float* hb   = hid + (size_t)b * N_ * M_;
  const int   mL    = L & 3;
  const bool  mOK   = (L < 4);          // rows 4..15 of A are zero padding

  v8f acc = {};
  for (int n0 = 0; n0 < N_; n0 += 4) {
    const int k0 = n0 + kOff;
    // A fragment: A[m=L][k] = hid[b, k0+k, L]  (zero for m >= 4)
    float ha0 = hb[(size_t)k0 * M_ + mL];
    float ha1 = hb[(size_t)(k0 + 1) * M_ + mL];
    v2f a;
    a.x = mOK ? ha0 : 0.f;
    a.y = mOK ? ha1 : 0.f;
    // B fragment: B[k][col=L] = xp[b, k0+k, c] recomputed on the fly
    float x0 = xrow[k0];
    float x1 = xrow[k0 + 1];
    float p0 = __sinf((float)k0 * div + phase);
    float p1 = __sinf((float)(k0 + 1) * div + phase);
    v2f bf;
    bf.x = x0 * invc + p0;
    bf.y = x1 * invc + p1;
    acc = __builtin_amdgcn_wmma_f32_16x16x4_f32(
        /*neg_a=*/false, a, /*neg_b=*/false, bf,
        /*c_mod=*/(short)0, acc, /*reuse_a=*/false, /*reuse_b=*/false);
  }

  // D layout: VGPR r, lanes 0-15 -> row M=r. We need rows m=0..3.
  if (lane < 16) {
    float pooled = 0.f;
    #pragma unroll
    for (int r = 0; r < 4; ++r)
      pooled += acc[r] / (hth[b * 4 + r] + LAMBDA_);
    // nan_to_num on the m-sum: NaN -> 0, +/-inf -> +/-FLT_MAX
    if (!(pooled == pooled)) pooled = 0.f;
    pooled = fminf(fmaxf(pooled, -3.4028234663852886e38f), 3.4028234663852886e38f);
    out[(size_t)b * C_ + c] = pooled;
  }
}

// ---------------------------------------------------------------------------
extern "C" void kernel_launch(void* const* d_in, const int* in_sizes, int n_in,
                              void* d_out, int out_size, void* d_ws, size_t ws_size,
                              hipStream_t stream) {
  const float* x = (const float*)d_in[0];   // (B, C, S, S)
  const float* W = (const float*)d_in[1];   // (M, C, 1)
  float* out = (float*)d_out;               // (B, 1, C)

  // Workspace: invn (B*C) | hid (B*N*M) | hth (B*M)
  float* invn = (float*)d_ws;
  float* hid  = invn + (size_t)B_ * C_;
  float* hth  = hid  + (size_t)B_ * N_ * M_;

  norm_kernel<<<dim3((B_ * C_) / 8), 256, 0, stream>>>(x, invn);
  hid_kernel<<<dim3(B_, (N_ + 255) / 256), 256, 0, stream>>>(x, W, invn, hid);
  hth_kernel<<<dim3(B_ * M_), 32, 0, stream>>>(hid, hth);
  htx_wmma_kernel<<<dim3(C_ / 16, B_), 32, 0, stream>>>(x, invn, hid, hth, out);
}